// DifferentialMaxtree_63187558859119
// MI455X (gfx1250) — compile-verified
//
#include <hip/hip_runtime.h>

// Problem constants (match reference)
#define B_   4
#define N_   8
#define H_   512
#define W_   512
#define C_   65536
#define CP1  (C_ + 1)
#define EPS_ 1e-10f

typedef __attribute__((ext_vector_type(2))) float v2f;
typedef __attribute__((ext_vector_type(8))) float v8f;

// Interleaved pointer-jumping state: one 8-byte node -> every random gather
// in the jump phase is a single global_load_b64 instead of two b32 gathers.
struct __align__(8) Node { float s; int p; };

// ---------------------------------------------------------------------------
// Kernel A: features -> logits (WMMA f32 16x16x4, K=17 padded to 20) ->
//           sigmoid -> w = diff*score ; writes {w, parent} nodes and the
//           per-(b,n) sentinel node {0, C}.
// Each wave handles 16 elements: lane L computes features of element (L&15);
// half-waves supply K0/K1 (lanes 0-15) and K2/K3 (lanes 16-31) of the A tile.
// B tile broadcasts weight[n] chunk across all 16 N columns, so every D
// column is the desired dot product.
// ---------------------------------------------------------------------------
__global__ __launch_bounds__(256) void k_weights(
    const float* __restrict__ diff, const float* __restrict__ attrs,
    const float* __restrict__ weight, const float* __restrict__ bias,
    const int* __restrict__ parent, Node* __restrict__ nodes)
{
    const int bn   = blockIdx.y;          // b*N + n
    const int n    = bn & (N_ - 1);
    const int lane = threadIdx.x & 31;    // wave32
    const int wave = threadIdx.x >> 5;
    const int m    = lane & 15;           // element slot within wave tile
    const int h    = lane >> 4;           // half-wave: 0 -> K0/K1, 1 -> K2/K3
    const int elem = blockIdx.x * 128 + wave * 16 + m;
    const long g   = (long)bn * C_ + elem;

    // ---- load 15 attrs, build 17 features (padded to 20) ----
    const float* ap = attrs + g * 15;
    float a[15];
#pragma unroll
    for (int i = 0; i < 15; ++i) a[i] = ap[i];

    float f[20];
    f[0] = a[0]; f[1] = a[1]; f[2] = a[2]; f[3] = a[3];   // bbox
    f[4] = __logf(a[4]);                                  // log(area)
#pragma unroll
    for (int t = 0; t < 9; ++t) {                         // signed-log tail
        float x  = a[6 + t];
        float sg = (x > 0.0f) ? 1.0f : ((x < 0.0f) ? -1.0f : 0.0f);
        f[5 + t] = __logf(fabsf(x) + EPS_) * sg;
    }
    f[14] = __builtin_sqrtf(a[7]) / (__builtin_sqrtf(a[6]) + EPS_); // lshape
    f[15] = __cosf(a[5]);
    f[16] = __sinf(a[5]);
    f[17] = 0.0f; f[18] = 0.0f; f[19] = 0.0f;

    // ---- per-n weight vector (uniform across block -> scalar loads) ----
    const float* wp = weight + n * 17;
    float wv[20];
#pragma unroll
    for (int k = 0; k < 17; ++k) wv[k] = wp[k];
    wv[17] = 0.0f; wv[18] = 0.0f; wv[19] = 0.0f;

    // ---- 5 chained V_WMMA_F32_16X16X4_F32, accumulating over K chunks ----
    v8f acc = {0.f, 0.f, 0.f, 0.f, 0.f, 0.f, 0.f, 0.f};
#pragma unroll
    for (int j = 0; j < 5; ++j) {
        float a0 = h ? f[4 * j + 2]  : f[4 * j + 0];
        float a1 = h ? f[4 * j + 3]  : f[4 * j + 1];
        float b0 = h ? wv[4 * j + 2] : wv[4 * j + 0];
        float b1 = h ? wv[4 * j + 3] : wv[4 * j + 1];
        v2f Av = {a0, a1};
        v2f Bv = {b0, b1};
        acc = __builtin_amdgcn_wmma_f32_16x16x4_f32(
            /*neg_a=*/false, Av, /*neg_b=*/false, Bv,
            /*c_mod=*/(short)0, acc, /*reuse_a=*/false, /*reuse_b=*/false);
    }

    // ---- stage the 16 logits through LDS (D row M=r -> acc[r]) ----
    // lane 0 holds elems 0..7 (lo half), lane 16 holds elems 8..15.
    __shared__ float stage[8][16];
    if (m == 0) {
#pragma unroll
        for (int r = 0; r < 8; ++r) stage[wave][h * 8 + r] = acc[r];
    }
    __builtin_amdgcn_wave_barrier();
    asm volatile("s_wait_dscnt 0" ::: "memory");   // same-wave DS ordering

    if (lane < 16) {
        float logit = stage[wave][m] + bias[n];
        float sc    = 1.0f / (1.0f + __expf(-logit));
        Node nd;
        nd.s = diff[g] * sc;
        nd.p = parent[g];
        nodes[(long)bn * CP1 + elem] = nd;         // single b64 store
    }

    // sentinel: {s=0, p=C} once per bn
    if (blockIdx.x == 0 && threadIdx.x == 0) {
        Node nd; nd.s = 0.0f; nd.p = C_;
        nodes[(long)bn * CP1 + C_] = nd;
    }
}

// ---------------------------------------------------------------------------
// Kernel B: one pointer-jumping round on interleaved nodes:
//   out[i] = { in[i].s + in[p].s , in[p].p }    (p = in[i].p)
// 3x b64 memory ops per element; all buffers (~34 MB) are L2-resident.
// ---------------------------------------------------------------------------
__global__ __launch_bounds__(256) void k_jump(
    const Node* __restrict__ in, Node* __restrict__ out)
{
    int j = blockIdx.x * 256 + threadIdx.x;
    if (j >= CP1) return;
    long base = (long)blockIdx.y * CP1;
    long t    = base + j;
    Node a = in[t];              // coalesced b64
    Node b = in[base + a.p];     // random b64 gather (L2)
    Node r; r.s = a.s + b.s; r.p = b.p;
    out[t] = r;                  // coalesced b64
}

// Final round: ptr output is dead -> write only the compact value array.
__global__ __launch_bounds__(256) void k_jump_final(
    const Node* __restrict__ in, float* __restrict__ val)
{
    int j = blockIdx.x * 256 + threadIdx.x;
    if (j >= CP1) return;
    long base = (long)blockIdx.y * CP1;
    long t    = base + j;
    Node a = in[t];
    val[t] = a.s + in[base + a.p].s;
}

// ---------------------------------------------------------------------------
// Kernel C: out[b,n,h,w] = val[bn, pix2cc[b,n,h,w]] / 10
// pix2cc/out are streamed with non-temporal hints so the L2 keeps the
// dense 8 MB value array hot for the random gathers.
// ---------------------------------------------------------------------------
__global__ __launch_bounds__(256) void k_gather(
    const int* __restrict__ pix2cc, const float* __restrict__ val,
    float* __restrict__ out)
{
    long bn  = blockIdx.y;
    long pix = (long)blockIdx.x * 256 + threadIdx.x;
    long g   = bn * (long)(H_ * W_) + pix;
    int  cc  = __builtin_nontemporal_load(pix2cc + g);
    float v  = val[bn * CP1 + cc];
    __builtin_nontemporal_store(v * 0.1f, out + g);
}

// ---------------------------------------------------------------------------
extern "C" void kernel_launch(void* const* d_in, const int* in_sizes, int n_in,
                              void* d_out, int out_size, void* d_ws, size_t ws_size,
                              hipStream_t stream)
{
    const float* diff   = (const float*)d_in[0];
    const float* attrs  = (const float*)d_in[1];
    const float* weight = (const float*)d_in[2];
    const float* bias   = (const float*)d_in[3];
    const int*   parent = (const int*)d_in[4];
    const int*   pix2cc = (const int*)d_in[5];
    float*       out    = (float*)d_out;

    const long NTOT = (long)B_ * N_ * CP1;   // 2,097,184 nodes per buffer
    Node* nodeA = (Node*)d_ws;
    Node* nodeB = nodeA + NTOT;
    // Final compact value array reuses nodeB's storage (nodeB is not read
    // during the final round, which consumes nodeA).
    float* val  = (float*)nodeB;

    // Phase 1: WMMA matvec + sigmoid -> node buffer A (+ sentinels)
    dim3 gA(C_ / 128, B_ * N_);
    k_weights<<<gA, 256, 0, stream>>>(diff, attrs, weight, bias, parent, nodeA);

    // Phase 2: 16 ping-pong jump rounds + 1 value-only final round (17 total)
    dim3 gJ((CP1 + 255) / 256, B_ * N_);
    Node* in  = nodeA;
    Node* out_ = nodeB;
    for (int r = 0; r < 16; ++r) {
        k_jump<<<gJ, 256, 0, stream>>>(in, out_);
        Node* t = in; in = out_; out_ = t;
    }
    // After 16 rounds (even count) state is back in nodeA; final round
    // writes the dense value array into the nodeB region.
    k_jump_final<<<gJ, 256, 0, stream>>>(in, val);

    // Phase 3: pixel gather, non-temporal streaming
    dim3 gG((H_ * W_) / 256, B_ * N_);
    k_gather<<<gG, 256, 0, stream>>>(pix2cc, val, out);
}